// GAT_4252017623767
// MI455X (gfx1250) — compile-verified
//
#include <hip/hip_runtime.h>
#include <hip/hip_bf16.h>

typedef float v2f __attribute__((ext_vector_type(2)));
typedef float v8f __attribute__((ext_vector_type(8)));

#define D_IN 128
#define NEG_SLOPE 0.2f
#define ENC_NEG_INF 0x007FFFFFu  // order-preserving encoding of -inf
#define SMALL_ND 16
#define STAT_CHUNK 4096
#define SCAT_CHUNK 512

__device__ __forceinline__ float leaky(float x) { return x >= 0.f ? x : NEG_SLOPE * x; }
// monotone f32 -> u32 key (so unsigned atomicMax == float max)
__device__ __forceinline__ unsigned encf(float f) {
    unsigned u = __float_as_uint(f);
    return (u & 0x80000000u) ? ~u : (u | 0x80000000u);
}
__device__ __forceinline__ float decf(unsigned k) {
    unsigned u = (k & 0x80000000u) ? (k & 0x7FFFFFFFu) : ~k;
    return __uint_as_float(u);
}

// ----------------------------- init --------------------------------
__global__ void fill_f32(float* __restrict__ p, float v, long n) {
    long i = (long)blockIdx.x * blockDim.x + threadIdx.x;
    long s = (long)gridDim.x * blockDim.x;
    for (; i < n; i += s) p[i] = v;
}
__global__ void fill_u32(unsigned* __restrict__ p, unsigned v, long n) {
    long i = (long)blockIdx.x * blockDim.x + threadIdx.x;
    long s = (long)gridDim.x * blockDim.x;
    for (; i < n; i += s) p[i] = v;
}

// ------------------- projection: z = x*W, el/er dots ----------------
// One wave computes a 16-row x 128-col strip of z via 8 accumulator tiles,
// K=128 swept in 32 steps of V_WMMA_F32_16X16X4_F32 (exact f32 math).
// A fragment (16x4) is loaded once per K-step and reused over 8 N-tiles.
// el/er per (row, head) reduced in-register from the C/D VGPR layout:
//   acc[t][r]: lanes 0-15 hold row m0+r, lanes 16-31 hold row m0+8+r, col = t*16+lane%16.
__global__ __launch_bounds__(256) void project_wmma(
    const float* __restrict__ X, const float* __restrict__ W,
    const float* __restrict__ AL, const float* __restrict__ AR,
    float* __restrict__ Z, float* __restrict__ EL, float* __restrict__ ER, int N)
{
    const int wave = threadIdx.x >> 5;
    const int lane = threadIdx.x & 31;
    const long m0 = ((long)blockIdx.x * 8 + wave) * 16;
    if (m0 >= N) return;                      // wave-uniform exit; live waves keep EXEC all-1s
    const int lr = lane & 15;
    const int lh = lane >> 4;                 // 0: K even pair / rows m0+r ; 1: K odd pair / rows m0+8+r

    long rowA = m0 + lr; if (rowA >= N) rowA = N - 1;   // clamp OOB loads, skip OOB stores
    const v2f* __restrict__ x2 = (const v2f*)(X + rowA * D_IN);

    float al[8], ar[8];
#pragma unroll
    for (int t = 0; t < 8; ++t) { al[t] = AL[t * 16 + lr]; ar[t] = AR[t * 16 + lr]; }

    v8f acc[8];
#pragma unroll
    for (int t = 0; t < 8; ++t) acc[t] = (v8f)0.f;

    for (int kb = 0; kb < 32; ++kb) {
        const int k = kb * 4 + lh * 2;        // A layout: lanes 0-15 K={0,1}, lanes 16-31 K={2,3}
        v2f a = x2[k >> 1];
#pragma unroll
        for (int t = 0; t < 8; ++t) {
            const int n = t * 16 + lr;        // B layout mirrors A: per-lane rows K, col = lane%16
            const float* wp = W + (n >> 6) * (D_IN * 64) + (n & 63); // W[h][k][f], h=n/64, f=n%64
            v2f b; b.x = wp[k * 64]; b.y = wp[(k + 1) * 64];
            acc[t] = __builtin_amdgcn_wmma_f32_16x16x4_f32(
                false, a, false, b, (short)0, acc[t], false, false);
        }
    }

#pragma unroll
    for (int r = 0; r < 8; ++r) {
        const long row = m0 + r + lh * 8;
        float sl0 = 0.f, sl1 = 0.f, sr0 = 0.f, sr1 = 0.f;   // head0: t<4 (n<64), head1: t>=4
#pragma unroll
        for (int t = 0; t < 8; ++t) {
            float z = acc[t][r];
            if (t < 4) { sl0 += z * al[t]; sr0 += z * ar[t]; }
            else       { sl1 += z * al[t]; sr1 += z * ar[t]; }
            if (row < N) Z[row * D_IN + t * 16 + lr] = z;
        }
#pragma unroll
        for (int off = 8; off >= 1; off >>= 1) {            // reduce within each 16-lane half
            sl0 += __shfl_xor(sl0, off, 32);
            sl1 += __shfl_xor(sl1, off, 32);
            sr0 += __shfl_xor(sr0, off, 32);
            sr1 += __shfl_xor(sr1, off, 32);
        }
        if (lr == 0 && row < N) {
            EL[row * 2 + 0] = sl0; EL[row * 2 + 1] = sl1;
            ER[row * 2 + 0] = sr0; ER[row * 2 + 1] = sr1;
        }
    }
}

// --------------------- edge pass 1: segment max ---------------------
__global__ __launch_bounds__(256) void edge_max(
    const float* __restrict__ ELs, const float* __restrict__ ERd,
    const int* __restrict__ src, const int* __restrict__ dst,
    unsigned* __restrict__ emax, int E)
{
    int i = blockIdx.x * blockDim.x + threadIdx.x;
    if (i >= E) return;
    int s = src[i], d = dst[i];
    float2 l = ((const float2*)ELs)[s];
    float2 r = ((const float2*)ERd)[d];
    atomicMax(&emax[2 * d + 0], encf(leaky(l.x + r.x)));
    atomicMax(&emax[2 * d + 1], encf(leaky(l.y + r.y)));
}

__global__ __launch_bounds__(256) void edge_max_small(
    const float* __restrict__ ELs, const float* __restrict__ ERd,
    const int* __restrict__ src, const int* __restrict__ dst,
    unsigned* __restrict__ emax, int E, int nd)
{
    __shared__ unsigned smax[SMALL_ND * 2];
    if (threadIdx.x < (unsigned)(nd * 2)) smax[threadIdx.x] = 0u;  // 0 < ENC_NEG_INF: merge-neutral
    __syncthreads();
    int base = blockIdx.x * STAT_CHUNK;
    int end = min(base + STAT_CHUNK, E);
    for (int i = base + threadIdx.x; i < end; i += 256) {
        int s = src[i], d = dst[i];
        float2 l = ((const float2*)ELs)[s];
        float2 r = ((const float2*)ERd)[d];
        atomicMax(&smax[2 * d + 0], encf(leaky(l.x + r.x)));   // ds_max_u32
        atomicMax(&smax[2 * d + 1], encf(leaky(l.y + r.y)));
    }
    __syncthreads();
    if (threadIdx.x < (unsigned)(nd * 2)) atomicMax(&emax[threadIdx.x], smax[threadIdx.x]);
}

// --------------------- edge pass 2: segment sum ---------------------
__global__ __launch_bounds__(256) void edge_sum(
    const float* __restrict__ ELs, const float* __restrict__ ERd,
    const unsigned* __restrict__ emax, float* __restrict__ denom,
    const int* __restrict__ src, const int* __restrict__ dst, int E)
{
    int i = blockIdx.x * blockDim.x + threadIdx.x;
    if (i >= E) return;
    int s = src[i], d = dst[i];
    float2 l = ((const float2*)ELs)[s];
    float2 r = ((const float2*)ERd)[d];
    atomicAdd(&denom[2 * d + 0], expf(leaky(l.x + r.x) - decf(emax[2 * d + 0])));
    atomicAdd(&denom[2 * d + 1], expf(leaky(l.y + r.y) - decf(emax[2 * d + 1])));
}

__global__ __launch_bounds__(256) void edge_sum_small(
    const float* __restrict__ ELs, const float* __restrict__ ERd,
    const unsigned* __restrict__ emax, float* __restrict__ denom,
    const int* __restrict__ src, const int* __restrict__ dst, int E, int nd)
{
    __shared__ float ssum[SMALL_ND * 2];
    if (threadIdx.x < (unsigned)(nd * 2)) ssum[threadIdx.x] = 0.f;
    __syncthreads();
    int base = blockIdx.x * STAT_CHUNK;
    int end = min(base + STAT_CHUNK, E);
    for (int i = base + threadIdx.x; i < end; i += 256) {
        int s = src[i], d = dst[i];
        float2 l = ((const float2*)ELs)[s];
        float2 r = ((const float2*)ERd)[d];
        atomicAdd(&ssum[2 * d + 0], expf(leaky(l.x + r.x) - decf(emax[2 * d + 0])));  // ds_add_f32
        atomicAdd(&ssum[2 * d + 1], expf(leaky(l.y + r.y) - decf(emax[2 * d + 1])));
    }
    __syncthreads();
    if (threadIdx.x < (unsigned)(nd * 2)) atomicAdd(&denom[threadIdx.x], ssum[threadIdx.x]);
}

// ------------- edge pass 3: alpha-weighted gather/scatter -----------
// One wave per edge: lane carries 4 consecutive cols (float4 gather, 4 f32 atomics).
__global__ __launch_bounds__(256) void edge_scatter(
    const float* __restrict__ Zs, const float* __restrict__ ELs, const float* __restrict__ ERd,
    const unsigned* __restrict__ emax, const float* __restrict__ denom,
    const int* __restrict__ src, const int* __restrict__ dst,
    float* __restrict__ out, int E)
{
    long gid = (long)blockIdx.x * blockDim.x + threadIdx.x;
    int lane = (int)(gid & 31);
    long e = gid >> 5;
    if (e >= E) return;
    int s = src[e], d = dst[e];
    int col = lane * 4;
    int h = lane >> 4;                                   // cols [0,64) head 0, [64,128) head 1
    float ee = expf(leaky(ELs[2 * s + h] + ERd[2 * d + h]) - decf(emax[2 * d + h]));
    float alpha = ee / (denom[2 * d + h] + 1e-9f);
    float4 z = *(const float4*)(Zs + (long)s * D_IN + col);
    float* o = out + (long)d * D_IN + col;
    atomicAdd(o + 0, z.x * alpha);
    atomicAdd(o + 1, z.y * alpha);
    atomicAdd(o + 2, z.z * alpha);
    atomicAdd(o + 3, z.w * alpha);
}

__global__ __launch_bounds__(256) void edge_scatter_small(
    const float* __restrict__ Zs, const float* __restrict__ ELs, const float* __restrict__ ERd,
    const unsigned* __restrict__ emax, const float* __restrict__ denom,
    const int* __restrict__ src, const int* __restrict__ dst,
    float* __restrict__ out, int E, int nd)
{
    __shared__ float sacc[SMALL_ND * D_IN];              // whole dst set staged in LDS
    for (int i = threadIdx.x; i < nd * D_IN; i += 256) sacc[i] = 0.f;
    __syncthreads();
    int wave = threadIdx.x >> 5, lane = threadIdx.x & 31;
    int col = lane * 4, h = lane >> 4;
    int base = blockIdx.x * SCAT_CHUNK;
    int nE = min(SCAT_CHUNK, E - base);
    for (int j = wave; j < nE; j += 8) {
        int e = base + j;
        int s = src[e], d = dst[e];
        float ee = expf(leaky(ELs[2 * s + h] + ERd[2 * d + h]) - decf(emax[2 * d + h]));
        float alpha = ee / (denom[2 * d + h] + 1e-9f);
        float4 z = *(const float4*)(Zs + (long)s * D_IN + col);
        float* o = &sacc[d * D_IN + col];
        atomicAdd(o + 0, z.x * alpha);
        atomicAdd(o + 1, z.y * alpha);
        atomicAdd(o + 2, z.z * alpha);
        atomicAdd(o + 3, z.w * alpha);
    }
    __syncthreads();
    for (int i = threadIdx.x; i < nd * D_IN; i += 256) atomicAdd(&out[i], sacc[i]);
}

// ------------------------------ host --------------------------------
extern "C" void kernel_launch(void* const* d_in, const int* in_sizes, int n_in,
                              void* d_out, int out_size, void* d_ws, size_t ws_size,
                              hipStream_t stream)
{
    const int N_USERS = 200000, N_ITEMS = 100000, N_CATES = 1000, N_RATES = 10;
    const int E_UI = 600000, E_IC = 300000, E_IR = 200000;

    const float* user_emb = (const float*)d_in[0];
    const float* item_emb = (const float*)d_in[1];
    const float* cate_emb = (const float*)d_in[2];
    const float* rate_emb = (const float*)d_in[3];
    const float* Wp       = (const float*)d_in[4];   // [2][128][64]
    const float* AL       = (const float*)d_in[5];   // [2][64] -> flat col index
    const float* AR       = (const float*)d_in[6];
    const int* iu_src = (const int*)d_in[7];  const int* iu_dst = (const int*)d_in[8];
    const int* ui_src = (const int*)d_in[9];  const int* ui_dst = (const int*)d_in[10];
    const int* ic_src = (const int*)d_in[11]; const int* ic_dst = (const int*)d_in[12];
    const int* ci_src = (const int*)d_in[13]; const int* ci_dst = (const int*)d_in[14];
    const int* ir_src = (const int*)d_in[15]; const int* ir_dst = (const int*)d_in[16];
    const int* ri_src = (const int*)d_in[17]; const int* ri_dst = (const int*)d_in[18];

    // ---- workspace carve-up (floats) ----
    float* ws = (float*)d_ws;
    size_t off = 0;
    float* z_user = ws + off; off += (size_t)N_USERS * D_IN;
    float* z_item = ws + off; off += (size_t)N_ITEMS * D_IN;
    float* z_cate = ws + off; off += (size_t)N_CATES * D_IN;
    float* z_rate = ws + off; off += (size_t)N_RATES * D_IN;
    float* el_user = ws + off; off += (size_t)N_USERS * 2;
    float* er_user = ws + off; off += (size_t)N_USERS * 2;
    float* el_item = ws + off; off += (size_t)N_ITEMS * 2;
    float* er_item = ws + off; off += (size_t)N_ITEMS * 2;
    float* el_cate = ws + off; off += (size_t)N_CATES * 2;
    float* er_cate = ws + off; off += (size_t)N_CATES * 2;
    float* el_rate = ws + off; off += (size_t)N_RATES * 2;
    float* er_rate = ws + off; off += (size_t)N_RATES * 2;

    const size_t stat_cnt = 2ul * (N_USERS + N_ITEMS + N_CATES + N_ITEMS + N_RATES + N_ITEMS);
    unsigned* emax_base = (unsigned*)(ws + off); off += stat_cnt;
    float*    den_base  = ws + off;              off += stat_cnt;

    unsigned* emax_iu = emax_base;
    unsigned* emax_ui = emax_iu + 2 * N_USERS;
    unsigned* emax_ic = emax_ui + 2 * N_ITEMS;
    unsigned* emax_ci = emax_ic + 2 * N_CATES;
    unsigned* emax_ir = emax_ci + 2 * N_ITEMS;
    unsigned* emax_ri = emax_ir + 2 * N_RATES;
    float* den_iu = den_base;
    float* den_ui = den_iu + 2 * N_USERS;
    float* den_ic = den_ui + 2 * N_ITEMS;
    float* den_ci = den_ic + 2 * N_CATES;
    float* den_ir = den_ci + 2 * N_ITEMS;
    float* den_ri = den_ir + 2 * N_RATES;

    float* out = (float*)d_out;
    float* out_user      = out;                                  // h_user
    float* out_item      = out + (size_t)N_USERS * D_IN;         // h_item
    float* out_cate      = out_item + (size_t)N_ITEMS * D_IN;    // h_cate_idx
    float* out_item_cate = out_cate + (size_t)N_CATES * D_IN;    // h_item_cate
    float* out_rate      = out_item_cate + (size_t)N_ITEMS * D_IN; // h_rate_idx
    float* out_item_rate = out_rate + (size_t)N_RATES * D_IN;    // h_item_rate

    // ---- re-init accumulators every call (graph replays reuse buffers) ----
    fill_f32<<<4096, 256, 0, stream>>>(out, 0.f, (long)out_size);
    fill_u32<<<512, 256, 0, stream>>>(emax_base, ENC_NEG_INF, (long)stat_cnt);
    fill_f32<<<512, 256, 0, stream>>>(den_base, 0.f, (long)stat_cnt);

    // ---- projections: z, el, er for each node set ----
    auto pgrid = [](int N) { return (N + 127) / 128; };
    project_wmma<<<pgrid(N_USERS), 256, 0, stream>>>(user_emb, Wp, AL, AR, z_user, el_user, er_user, N_USERS);
    project_wmma<<<pgrid(N_ITEMS), 256, 0, stream>>>(item_emb, Wp, AL, AR, z_item, el_item, er_item, N_ITEMS);
    project_wmma<<<pgrid(N_CATES), 256, 0, stream>>>(cate_emb, Wp, AL, AR, z_cate, el_cate, er_cate, N_CATES);
    project_wmma<<<pgrid(N_RATES), 256, 0, stream>>>(rate_emb, Wp, AL, AR, z_rate, el_rate, er_rate, N_RATES);

    // ---- six edge types ----
    struct Et {
        const float* zs; const float* els; const float* erd;
        unsigned* emax; float* denom;
        const int* src; const int* dst; float* o; int E; int Nd;
    };
    Et ets[6] = {
        { z_item, el_item, er_user, emax_iu, den_iu, iu_src, iu_dst, out_user,      E_UI, N_USERS },
        { z_user, el_user, er_item, emax_ui, den_ui, ui_src, ui_dst, out_item,      E_UI, N_ITEMS },
        { z_item, el_item, er_cate, emax_ic, den_ic, ic_src, ic_dst, out_cate,      E_IC, N_CATES },
        { z_cate, el_cate, er_item, emax_ci, den_ci, ci_src, ci_dst, out_item_cate, E_IC, N_ITEMS },
        { z_item, el_item, er_rate, emax_ir, den_ir, ir_src, ir_dst, out_rate,      E_IR, N_RATES },
        { z_rate, el_rate, er_item, emax_ri, den_ri, ri_src, ri_dst, out_item_rate, E_IR, N_ITEMS },
    };

    for (int i = 0; i < 6; ++i) {
        const Et& c = ets[i];
        if (c.Nd <= SMALL_ND) {   // tiny dst set: stage in LDS, merge once per block
            int gs = (c.E + STAT_CHUNK - 1) / STAT_CHUNK;
            edge_max_small<<<gs, 256, 0, stream>>>(c.els, c.erd, c.src, c.dst, c.emax, c.E, c.Nd);
            edge_sum_small<<<gs, 256, 0, stream>>>(c.els, c.erd, c.emax, c.denom, c.src, c.dst, c.E, c.Nd);
            int gsc = (c.E + SCAT_CHUNK - 1) / SCAT_CHUNK;
            edge_scatter_small<<<gsc, 256, 0, stream>>>(c.zs, c.els, c.erd, c.emax, c.denom,
                                                        c.src, c.dst, c.o, c.E, c.Nd);
        } else {
            int ge = (c.E + 255) / 256;
            edge_max<<<ge, 256, 0, stream>>>(c.els, c.erd, c.src, c.dst, c.emax, c.E);
            edge_sum<<<ge, 256, 0, stream>>>(c.els, c.erd, c.emax, c.denom, c.src, c.dst, c.E);
            int gsc = (int)(((long)c.E * 32 + 255) / 256);
            edge_scatter<<<gsc, 256, 0, stream>>>(c.zs, c.els, c.erd, c.emax, c.denom,
                                                  c.src, c.dst, c.o, c.E);
        }
    }
}